// TensoRFGrid_23373212025334
// MI455X (gfx1250) — compile-verified
//
#include <hip/hip_runtime.h>

typedef __attribute__((ext_vector_type(2))) float v2f;
typedef __attribute__((ext_vector_type(8))) float v8f;

#define R_RANK   48
#define DIMV     300
#define PLSZ     (DIMV * DIMV)    // 90000
#define C_OUT    27
#define K_FEAT   144              // 3 * R_RANK
#define NPAD     40               // padded f_vec row stride (floats), bank-friendly
#define FSTRIDE  148              // padded feat row stride (floats)
#define WAVES_PB 4
#define PTS_PW   16
#define PTS_PB   (WAVES_PB * PTS_PW)   // 64
#define THREADS  (WAVES_PB * 32)       // 128

// Equivalent to the reference clamp for coord in [0,1]: i0 in [0, DIMV-2],
// i1 = i0+1 always in-bounds, f = ih - i0 (f hits 1.0 exactly at the top edge).
__device__ __forceinline__ void axis_setup(float coord, int& i0, float& f) {
    float ih = coord * (float)(DIMV - 1);   // align_corners=true over [0,1]
    int i = (int)floorf(ih);
    i = i < 0 ? 0 : (i > DIMV - 2 ? DIMV - 2 : i);
    i0 = i;
    f = ih - (float)i;
}

// 8-byte paired load at 4-byte alignment (DWORD alignment mode allows
// 4-aligned B64 on gfx1250); memcpy keeps clang's assumed align at 4.
__device__ __forceinline__ v2f ld2(const float* __restrict__ p) {
    v2f r;
    __builtin_memcpy(&r, p, sizeof(r));
    return r;
}

// Bilinear tap: two paired loads (rows i0, i0+1), lerp fj then fi.
__device__ __forceinline__ float bilin(const float* __restrict__ at,
                                       float fi, float fj) {
    v2f t = ld2(at);
    v2f b = ld2(at + DIMV);
    float top = t.x + (t.y - t.x) * fj;
    float bot = b.x + (b.y - b.x) * fj;
    return top + (bot - top) * fi;
}

__device__ __forceinline__ float lerp1(const float* __restrict__ at, float f) {
    v2f v = ld2(at);
    return v.x + (v.y - v.x) * f;
}

__global__ __launch_bounds__(THREADS)
void tensorf_feat_gemm(const float* __restrict__ xyz,
                       const float* __restrict__ xy_plane,
                       const float* __restrict__ xz_plane,
                       const float* __restrict__ yz_plane,
                       const float* __restrict__ x_vec,
                       const float* __restrict__ y_vec,
                       const float* __restrict__ z_vec,
                       const float* __restrict__ f_vec,
                       float* __restrict__ out,
                       int n) {
    __shared__ float fv[K_FEAT * NPAD];                     // 144*40*4 = 23040 B
    __shared__ float feat[WAVES_PB][PTS_PW * FSTRIDE];      // 4*16*148*4 = 37888 B

    const int tid  = threadIdx.x;
    const int wave = tid >> 5;
    const int lane = tid & 31;

    // Stage f_vec [144 x 27] into LDS padded to 40 columns (zeros beyond 27).
    for (int i = tid; i < K_FEAT * NPAD; i += THREADS) {
        int k = i / NPAD, c = i - k * NPAD;
        fv[i] = (c < C_OUT) ? f_vec[k * C_OUT + c] : 0.0f;
    }

    const int p    = lane & 15;     // point within the wave's 16-point tile
    const int half = lane >> 4;     // channel half: 0 -> r 0..23, 1 -> r 24..47

    const int idx  = blockIdx.x * PTS_PB + wave * PTS_PW + p;
    const int lidx = idx < n ? idx : (n - 1);

    const float x = xyz[lidx * 3 + 0];
    const float y = xyz[lidx * 3 + 1];
    const float z = xyz[lidx * 3 + 2];

    int ix0, iy0, iz0;
    float fx, fy, fz;
    axis_setup(x, ix0, fx);
    axis_setup(y, iy0, fy);
    axis_setup(z, iz0, fz);

    // Per-plane 2D tap offsets (row i0, col j0); channel stride is PLSZ.
    const int rb = half * 24;
    const float* pxy = xy_plane + rb * PLSZ + (ix0 * DIMV + iy0);
    const float* pxz = xz_plane + rb * PLSZ + (ix0 * DIMV + iz0);
    const float* pyz = yz_plane + rb * PLSZ + (iy0 * DIMV + iz0);
    const float* vx  = x_vec + rb * DIMV + ix0;
    const float* vy  = y_vec + rb * DIMV + iy0;
    const float* vz  = z_vec + rb * DIMV + iz0;

    float* frow = &feat[wave][0];
    float* fdst = frow + p * FSTRIDE + rb;

    #pragma unroll 4
    for (int rr = 0; rr < 24; ++rr) {
        // xy-plane at (x, y) scaled by z_vec
        fdst[rr]      = bilin(pxy, fx, fy) * lerp1(vz, fz);
        // xz-plane at (x, z) scaled by y_vec
        fdst[48 + rr] = bilin(pxz, fx, fz) * lerp1(vy, fy);
        // yz-plane at (y, z) scaled by x_vec
        fdst[96 + rr] = bilin(pyz, fy, fz) * lerp1(vx, fx);
        pxy += PLSZ; pxz += PLSZ; pyz += PLSZ;
        vx += DIMV; vy += DIMV; vz += DIMV;
    }

    __syncthreads();

    // GEMM: feat[16 x 144] @ f_vec[144 x 27] via V_WMMA_F32_16X16X4_F32.
    // A (16x4 f32): lanes 0-15 hold (M=lane, K=k0+0 / k0+1); lanes 16-31 hold K=k0+2 / k0+3.
    // B (4x16 f32): lanes 0-15 hold (K=k0+0/+1, N=lane); lanes 16-31 hold K=k0+2/+3.
    v8f c0 = {};
    v8f c1 = {};
    const int m    = lane & 15;
    const int nb   = lane & 15;
    const int koff = half * 2;

    #pragma unroll 4
    for (int k0 = 0; k0 < K_FEAT; k0 += 4) {
        v2f a, b0, b1;
        a.x  = frow[m * FSTRIDE + k0 + koff];
        a.y  = frow[m * FSTRIDE + k0 + koff + 1];
        b0.x = fv[(k0 + koff)     * NPAD + nb];
        b0.y = fv[(k0 + koff + 1) * NPAD + nb];
        b1.x = fv[(k0 + koff)     * NPAD + nb + 16];
        b1.y = fv[(k0 + koff + 1) * NPAD + nb + 16];
        c0 = __builtin_amdgcn_wmma_f32_16x16x4_f32(false, a, false, b0,
                                                   (short)0, c0, false, false);
        c1 = __builtin_amdgcn_wmma_f32_16x16x4_f32(false, a, false, b1,
                                                   (short)0, c1, false, false);
    }

    // D layout: VGPR j, lanes 0-15 -> (M=j, N=lane); lanes 16-31 -> (M=j+8, N=lane-16)
    const int obase = blockIdx.x * PTS_PB + wave * PTS_PW;
    #pragma unroll
    for (int j = 0; j < 8; ++j) {
        const int mm   = j + half * 8;
        const int oidx = obase + mm;
        if (oidx < n) {
            out[oidx * C_OUT + nb] = c0[j];       // nb < 16 < 27 always valid
            const int n1 = nb + 16;
            if (n1 < C_OUT)
                out[oidx * C_OUT + n1] = c1[j];
        }
    }
}

extern "C" void kernel_launch(void* const* d_in, const int* in_sizes, int n_in,
                              void* d_out, int out_size, void* d_ws, size_t ws_size,
                              hipStream_t stream) {
    const float* xyz      = (const float*)d_in[0];
    const float* xy_plane = (const float*)d_in[1];
    const float* xz_plane = (const float*)d_in[2];
    const float* yz_plane = (const float*)d_in[3];
    const float* x_vec    = (const float*)d_in[4];
    const float* y_vec    = (const float*)d_in[5];
    const float* z_vec    = (const float*)d_in[6];
    const float* f_vec    = (const float*)d_in[7];
    float* out            = (float*)d_out;

    const int n = in_sizes[0] / 3;
    const int blocks = (n + PTS_PB - 1) / PTS_PB;
    tensorf_feat_gemm<<<blocks, THREADS, 0, stream>>>(
        xyz, xy_plane, xz_plane, yz_plane, x_vec, y_vec, z_vec, f_vec, out, n);
}